// CropAndResize_592705487147
// MI455X (gfx1250) — compile-verified
//
#include <hip/hip_runtime.h>

// CDNA5 / gfx1250 crop-and-resize via separable bilinear interpolation as
// two small matmuls on V_WMMA_F32_16X16X4_F32.
//
//   crop[c] (14x14) = Wy (14x28) @ P[c] (28x28) @ Wx^T (28x14)     per box
//
// padded to 16x32 @ 32x32 @ 32x16. P[c] = gathered patch of the 28 source
// rows (top/bot per y) x 28 source cols (left/right per x); identical memory
// footprint to the direct 4-tap gather (784 taps/channel), so the WMMA path
// is free on the memory side (HBM-gather bound: ~1 GB moved, ~45us floor at
// 23.3 TB/s). Out-of-range samples handled by zeroing weights (EXTRAP==0).
//
// Round-2 refinements:
//  * per-wave LDS sync (s_wait_dscnt 0 + compiler fence) instead of block
//    barriers -- sP/sT are wave-private; CDNA5 LDS ops are in-order per wave.
//  * column-major patch (pitch 34) so WMMA B-fragments are contiguous
//    8B-aligned float2 -> ds_load_b64 into even-aligned VGPR pairs.
//  * non-temporal output stores: keep the 192MB L2 for image-gather reuse
//    (~125 boxes share each 62MB image), stream the write-once output.

typedef __attribute__((ext_vector_type(2))) float v2f;
typedef __attribute__((ext_vector_type(8))) float v8f;

#define CROP    14
#define IMG_H   200
#define IMG_W   304
#define IMG_C   256

// Wave-local LDS fence: all prior DS stores of this wave complete + compiler
// may not reorder memory ops across it. Sufficient because sP/sT are private
// to the wave and CDNA5 processes a wave's LDS instructions in order.
__device__ __forceinline__ void wave_lds_fence() {
    asm volatile("s_wait_dscnt 0x0" ::: "memory");
}

__global__ __launch_bounds__(256)
void crop_resize_wmma(const float* __restrict__ img,
                      const float* __restrict__ boxes,
                      const int*   __restrict__ bindex,
                      float*       __restrict__ out)
{
    // Weight matrices, padded with zeros (rows 14,15 and slots 28..31 zero).
    __shared__ float sWy [16][32];     // A1:  Wy padded   (y  x rowslot)
    __shared__ float sWxT[32][16];     // B2:  Wx^T padded (colslot x x)
    __shared__ int   sRowOff[32];      // rowslot -> row*IMG_W
    __shared__ int   sCol[32];         // colslot -> col
    __shared__ float sPc[8][32][34];   // per-wave patch, COLUMN-major: [colslot][rowslot]
    __shared__ float sT [8][16][34];   // per-wave T = Wy@P staging (C/D -> A re-fragment)

    const int n   = blockIdx.x;
    const int tid = threadIdx.x;

    // ---- zero-init weight/index tables --------------------------------
    ((float*)sWy )[tid]       = 0.0f;
    ((float*)sWy )[tid + 256] = 0.0f;
    ((float*)sWxT)[tid]       = 0.0f;
    ((float*)sWxT)[tid + 256] = 0.0f;
    if (tid < 32) { sRowOff[tid] = 0; sCol[tid] = 0; }
    __syncthreads();

    // ---- per-box sampling coordinates + lerp weights ------------------
    const float y1 = boxes[n * 4 + 0];
    const float x1 = boxes[n * 4 + 1];
    const float y2 = boxes[n * 4 + 2];
    const float x2 = boxes[n * 4 + 3];

    if (tid < CROP) {                       // y coordinates (threads 0..13)
        const float stepY = (y2 - y1) * (float)(IMG_H - 1) / (float)(CROP - 1);
        const float iny   = y1 * (float)(IMG_H - 1) + (float)tid * stepY;
        const bool  valid = (iny >= 0.0f) && (iny <= (float)(IMG_H - 1));
        const float tf    = floorf(iny);
        const float yl    = iny - tf;
        const int   top   = (int)fminf(fmaxf(tf, 0.0f),         (float)(IMG_H - 1));
        const int   bot   = (int)fminf(fmaxf(ceilf(iny), 0.0f), (float)(IMG_H - 1));
        sRowOff[tid]        = top * IMG_W;
        sRowOff[CROP + tid] = bot * IMG_W;
        sWy[tid][tid]        = valid ? (1.0f - yl) : 0.0f;   // top weight
        sWy[tid][CROP + tid] = valid ? yl          : 0.0f;   // bottom weight
    } else if (tid >= 16 && tid < 16 + CROP) {  // x coordinates (threads 16..29)
        const int   t     = tid - 16;
        const float stepX = (x2 - x1) * (float)(IMG_W - 1) / (float)(CROP - 1);
        const float inx   = x1 * (float)(IMG_W - 1) + (float)t * stepX;
        const bool  valid = (inx >= 0.0f) && (inx <= (float)(IMG_W - 1));
        const float lf    = floorf(inx);
        const float xl    = inx - lf;
        const int   left  = (int)fminf(fmaxf(lf, 0.0f),         (float)(IMG_W - 1));
        const int   right = (int)fminf(fmaxf(ceilf(inx), 0.0f), (float)(IMG_W - 1));
        sCol[t]        = left;
        sCol[CROP + t] = right;
        sWxT[t][t]        = valid ? (1.0f - xl) : 0.0f;      // left weight
        sWxT[CROP + t][t] = valid ? xl          : 0.0f;      // right weight
    }
    __syncthreads();

    // ---- per-wave constant fragments ----------------------------------
    const int lane = tid & 31;
    const int w    = tid >> 5;          // wave id 0..7 -> channels w*32 .. w*32+31
    const int m    = lane & 15;
    const int koff = (lane >= 16) ? 2 : 0;   // A/B layout: VGPR0=K{0,2}, VGPR1=K{1,3}

    v2f aW[8], bX[8];
    #pragma unroll
    for (int k = 0; k < 8; ++k) {
        // A-matrix fragment of Wy (16x32): contiguous 8B-aligned pair
        aW[k] = *(const v2f*)&sWy[m][4 * k + koff];
        // B-matrix fragment of Wx^T (32x16): rows differ -> scalar picks (one-time)
        bX[k].x = sWxT[4 * k + koff + 0][m];
        bX[k].y = sWxT[4 * k + koff + 1][m];
    }

    int rOff[32];
    #pragma unroll
    for (int j = 0; j < 32; ++j) rOff[j] = sRowOff[j];
    const int myCol = sCol[lane];

    const float* plane0 = img + (size_t)bindex[n] * IMG_C * IMG_H * IMG_W;

    // ---- channel loop: 8 waves x 32 channels each (waves run decoupled) ----
    for (int i = 0; i < 32; ++i) {
        const int    c     = w * 32 + i;
        const float* plane = plane0 + (size_t)c * (IMG_H * IMG_W);

        // gather padded patch, column-major: lane = column slot, j = row slot
        #pragma unroll
        for (int j = 0; j < 32; ++j) {
            float v = 0.0f;
            if (j < 28 && lane < 28) v = plane[rOff[j] + myCol];
            sPc[w][lane][j] = v;
        }
        wave_lds_fence();

        // T(16x32) = Wy(16x32) @ P(32x32): two 16-col panels, 8 K-chunks each
        v8f acc0 = {0.f,0.f,0.f,0.f,0.f,0.f,0.f,0.f};
        v8f acc1 = {0.f,0.f,0.f,0.f,0.f,0.f,0.f,0.f};
        #pragma unroll
        for (int k = 0; k < 8; ++k) {
            // B fragments of P: rows 4k+koff,4k+koff+1 at col m / 16+m
            // column-major + pitch 34 -> contiguous 8B-aligned -> ds_load_b64
            v2f pb0 = *(const v2f*)&sPc[w][m]     [4 * k + koff];
            v2f pb1 = *(const v2f*)&sPc[w][16 + m][4 * k + koff];
            acc0 = __builtin_amdgcn_wmma_f32_16x16x4_f32(false, aW[k], false, pb0,
                                                         (short)0, acc0, false, false);
            acc1 = __builtin_amdgcn_wmma_f32_16x16x4_f32(false, aW[k], false, pb1,
                                                         (short)0, acc1, false, false);
        }

        // stage T through LDS to re-fragment C/D layout -> A layout
        #pragma unroll
        for (int v = 0; v < 8; ++v) {
            const int row = (lane >= 16) ? (v + 8) : v;
            sT[w][row][m]      = acc0[v];
            sT[w][row][16 + m] = acc1[v];
        }
        wave_lds_fence();

        // Out(16x16) = T(16x32) @ WxT(32x16)
        v8f o = {0.f,0.f,0.f,0.f,0.f,0.f,0.f,0.f};
        #pragma unroll
        for (int k = 0; k < 8; ++k) {
            v2f tf = *(const v2f*)&sT[w][m][4 * k + koff];   // 8B-aligned pair
            o = __builtin_amdgcn_wmma_f32_16x16x4_f32(false, tf, false, bX[k],
                                                      (short)0, o, false, false);
        }

        // store crop[y][x] non-temporally (write-once stream; keep L2 for image)
        if (m < CROP) {
            float* op = out + ((size_t)n * IMG_C + c) * (CROP * CROP) + m;
            #pragma unroll
            for (int v = 0; v < 8; ++v) {
                const int row = (lane >= 16) ? (v + 8) : v;
                if (row < CROP) __builtin_nontemporal_store(o[v], op + row * CROP);
            }
        }
        // next iteration's sP/sT writes are ordered behind this iteration's
        // reads by program order + the fences (per-wave in-order LDS).
    }
}

extern "C" void kernel_launch(void* const* d_in, const int* in_sizes, int n_in,
                              void* d_out, int out_size, void* d_ws, size_t ws_size,
                              hipStream_t stream) {
    (void)n_in; (void)out_size; (void)d_ws; (void)ws_size;
    const float* img   = (const float*)d_in[0];
    const float* boxes = (const float*)d_in[1];
    const int*   bidx  = (const int*)d_in[2];
    float*       out   = (float*)d_out;
    const int nBoxes   = in_sizes[2];      // 1000 boxes, one block per box
    hipLaunchKernelGGL(crop_resize_wmma, dim3(nBoxes), dim3(256), 0, stream,
                       img, boxes, bidx, out);
}